// SNNPolicy_12859132084311
// MI455X (gfx1250) — compile-verified
//
#include <hip/hip_runtime.h>
#include <hip/hip_bf16.h>

// ---------------------------------------------------------------------------
// Fused SNN policy forward on MI455X (gfx1250, wave32).
//   kernel 1: zero global spike-sum scratch
//   kernel 2: fused 16-timestep LIF network, f32 WMMA (v_wmma_f32_16x16x4_f32)
//   kernel 3: eligibility-trace epilogue from per-timestep batch spike sums
// ---------------------------------------------------------------------------

typedef float v2f __attribute__((ext_vector_type(2)));
typedef float v8f __attribute__((ext_vector_type(8)));

#define WMMA_F32(a, b, c) \
  __builtin_amdgcn_wmma_f32_16x16x4_f32(false, (a), false, (b), (short)0, (c), false, false)

namespace snn {
constexpr int   Bsz  = 65536;
constexpr int   S    = 25;
constexpr int   H    = 64;
constexpr int   A    = 4;
constexpr int   T    = 16;
constexpr float BETA = 0.95f;
constexpr float THR  = 1.0f;
constexpr float ELIG_DECAY = 0.95f;
constexpr float C_STDP     = 0.008f - 0.01f;   // A_PLUS - A_MINUS

constexpr int WAVES = 4;                // 4 waves x 16 rows = 64 rows / block
constexpr int BLOCK = WAVES * 32;
constexpr int ROWS_PER_BLOCK = WAVES * 16;
constexpr int KP   = 28;                // input K padded 25 -> 28 (7 k-chunks of 4)
constexpr int LPAD = 68;                // 64-col tiles padded: banks (4*row+col)%64, conflict-free
}  // namespace snn

__global__ void snn_zero_sums_kernel(float* __restrict__ gsum2, float* __restrict__ gsum3) {
  int i = blockIdx.x * blockDim.x + threadIdx.x;
  if (i < snn::T * snn::H) gsum2[i] = 0.f;
  if (i < snn::T * snn::A) gsum3[i] = 0.f;
}

__global__ __launch_bounds__(snn::BLOCK)
void snn_forward_kernel(const float* __restrict__ x,
                        const float* __restrict__ W1, const float* __restrict__ b1,
                        const float* __restrict__ W2, const float* __restrict__ b2,
                        const float* __restrict__ W3, const float* __restrict__ b3,
                        float* __restrict__ probs,
                        float* __restrict__ gsum2, float* __restrict__ gsum3) {
  using namespace snn;
  __shared__ float W1p[H][KP];          // W1 row-major, K zero-padded to 28
  __shared__ float W2p[H][LPAD];        // W2 row-major, padded stride
  __shared__ float W3p[16][LPAD];       // W3 rows 0..3, rows 4..15 zero
  __shared__ float b1s[H], b2s[H], b3s[16];
  __shared__ float stage[WAVES][16][LPAD];   // per-wave spike / input staging
  __shared__ float sum2[T][H];          // per-block per-timestep s2 column sums
  __shared__ float sum3[T][A];          // per-block per-timestep s3 column sums

  const int tid  = threadIdx.x;
  const int w    = tid >> 5;
  const int lane = tid & 31;
  const int hi   = lane >> 4;           // which 16-lane half
  const int lq   = lane & 15;
  const int rowBase = blockIdx.x * ROWS_PER_BLOCK + w * 16;

  // ---- cooperative LDS fill -------------------------------------------------
  for (int i = tid; i < H * KP; i += BLOCK) {
    int r = i / KP, c = i % KP;
    W1p[r][c] = (c < S) ? W1[r * S + c] : 0.f;
  }
  for (int i = tid; i < H * H; i += BLOCK) {
    W2p[i >> 6][i & 63] = W2[i];
  }
  for (int i = tid; i < 16 * H; i += BLOCK) {
    int r = i >> 6, c = i & 63;
    W3p[r][c] = (r < A) ? W3[r * H + c] : 0.f;
  }
  if (tid < H) { b1s[tid] = b1[tid]; b2s[tid] = b2[tid]; }
  if (tid < 16) b3s[tid] = (tid < A) ? b3[tid] : 0.f;
  for (int i = tid; i < T * H; i += BLOCK) (&sum2[0][0])[i] = 0.f;
  if (tid < T * A) (&sum3[0][0])[tid] = 0.f;
  __syncthreads();

  // ---- stage this wave's 16 input rows (K padded with zeros) ----------------
  for (int i = lane; i < 16 * KP; i += 32) {
    int r = i / KP, c = i % KP;
    stage[w][r][c] = (c < S) ? x[(size_t)(rowBase + r) * S + c] : 0.f;
  }

  // ---- cur1 = x @ W1^T + b1  (timestep-invariant, kept in registers) --------
  v8f cur1[4];
  #pragma unroll
  for (int nt = 0; nt < 4; ++nt) cur1[nt] = v8f{0.f,0.f,0.f,0.f,0.f,0.f,0.f,0.f};
  #pragma unroll
  for (int kc = 0; kc < KP / 4; ++kc) {
    v2f a = *(const v2f*)&stage[w][lq][4 * kc + 2 * hi];       // A: M=lq, K striped
    #pragma unroll
    for (int nt = 0; nt < 4; ++nt) {
      v2f b = *(const v2f*)&W1p[lq + 16 * nt][4 * kc + 2 * hi];  // B[k][n]=W1[n][k]
      cur1[nt] = WMMA_F32(a, b, cur1[nt]);
    }
  }
  #pragma unroll
  for (int nt = 0; nt < 4; ++nt) {
    float bb = b1s[lq + 16 * nt];
    #pragma unroll
    for (int r = 0; r < 8; ++r) cur1[nt][r] += bb;
  }

  // ---- LIF state ------------------------------------------------------------
  v8f m1[4], m2t[4], m3t, accv;
  #pragma unroll
  for (int nt = 0; nt < 4; ++nt) {
    m1[nt]  = v8f{0.f,0.f,0.f,0.f,0.f,0.f,0.f,0.f};
    m2t[nt] = v8f{0.f,0.f,0.f,0.f,0.f,0.f,0.f,0.f};
  }
  m3t  = v8f{0.f,0.f,0.f,0.f,0.f,0.f,0.f,0.f};
  accv = v8f{0.f,0.f,0.f,0.f,0.f,0.f,0.f,0.f};
  float biasW2[4];
  #pragma unroll
  for (int nt = 0; nt < 4; ++nt) biasW2[nt] = b2s[lq + 16 * nt];
  const float b3v = b3s[lq];   // zero for lq >= 4

  // ---- timestep loop --------------------------------------------------------
  for (int t = 0; t < T; ++t) {
    // layer 1: m1 = beta*m1 + cur1; spike; stage s1 (C-layout -> LDS row-major)
    #pragma unroll
    for (int nt = 0; nt < 4; ++nt) {
      #pragma unroll
      for (int r = 0; r < 8; ++r) {
        float m = BETA * m1[nt][r] + cur1[nt][r];
        float s = (m > THR) ? 1.f : 0.f;
        m1[nt][r] = m - s;
        stage[w][8 * hi + r][lq + 16 * nt] = s;
      }
    }

    // layer 2: m2 = beta*m2 + b2 + s1 @ W2^T   (16 chained K=4 f32 WMMAs / tile)
    #pragma unroll
    for (int nt = 0; nt < 4; ++nt) {
      #pragma unroll
      for (int r = 0; r < 8; ++r) m2t[nt][r] = BETA * m2t[nt][r] + biasW2[nt];
    }
    #pragma unroll
    for (int kc = 0; kc < 16; ++kc) {
      v2f a = *(const v2f*)&stage[w][lq][4 * kc + 2 * hi];
      #pragma unroll
      for (int nt = 0; nt < 4; ++nt) {
        v2f b = *(const v2f*)&W2p[lq + 16 * nt][4 * kc + 2 * hi];
        m2t[nt] = WMMA_F32(a, b, m2t[nt]);
      }
    }

    // s2 spikes: restage + per-column spike counts for pre-synaptic mean
    #pragma unroll
    for (int nt = 0; nt < 4; ++nt) {
      float colSum = 0.f;
      #pragma unroll
      for (int r = 0; r < 8; ++r) {
        float m = m2t[nt][r];
        float s = (m > THR) ? 1.f : 0.f;
        m2t[nt][r] = m - s;
        stage[w][8 * hi + r][lq + 16 * nt] = s;
        colSum += s;
      }
      colSum += __shfl_xor(colSum, 16, 32);          // combine row halves
      if (hi == 0) atomicAdd(&sum2[t][lq + 16 * nt], colSum);   // ds_add_f32
    }

    // layer 3: m3 = beta*m3 + b3 + s2 @ W3^T   (cols 4..15 stay exactly 0)
    #pragma unroll
    for (int r = 0; r < 8; ++r) m3t[r] = BETA * m3t[r] + b3v;
    #pragma unroll
    for (int kc = 0; kc < 16; ++kc) {
      v2f a = *(const v2f*)&stage[w][lq][4 * kc + 2 * hi];
      v2f b = *(const v2f*)&W3p[lq][4 * kc + 2 * hi];
      m3t = WMMA_F32(a, b, m3t);
    }
    float colSum3 = 0.f;
    #pragma unroll
    for (int r = 0; r < 8; ++r) {
      float m = m3t[r];
      float s = (m > THR) ? 1.f : 0.f;
      m3t[r] = m - s;
      accv[r] += s;
      colSum3 += s;
    }
    colSum3 += __shfl_xor(colSum3, 16, 32);
    if (hi == 0 && lq < A) atomicAdd(&sum3[t][lq], colSum3);
  }

  // ---- stage spike accumulator for per-row softmax --------------------------
  #pragma unroll
  for (int r = 0; r < 8; ++r) stage[w][8 * hi + r][lq] = accv[r];
  __syncthreads();   // also orders all sum2/sum3 LDS atomics before the flush

  if (hi == 0) {     // lanes 0..15: one batch row each
    const float invT = 1.f / (float)T;
    float v0 = stage[w][lq][0] * invT;
    float v1 = stage[w][lq][1] * invT;
    float v2 = stage[w][lq][2] * invT;
    float v3 = stage[w][lq][3] * invT;
    float mx = fmaxf(fmaxf(v0, v1), fmaxf(v2, v3));
    float e0 = __expf(v0 - mx), e1 = __expf(v1 - mx);
    float e2 = __expf(v2 - mx), e3 = __expf(v3 - mx);
    float inv = 1.f / (e0 + e1 + e2 + e3);
    float4 o = make_float4(e0 * inv, e1 * inv, e2 * inv, e3 * inv);
    *(float4*)&probs[(size_t)(rowBase + lq) * 4] = o;
  }

  // ---- flush per-block spike sums to global ---------------------------------
  for (int i = tid; i < T * H; i += BLOCK) atomicAdd(&gsum2[i], (&sum2[0][0])[i]);
  if (tid < T * A) atomicAdd(&gsum3[tid], (&sum3[0][0])[tid]);
}

__global__ void snn_elig_kernel(const float* __restrict__ elig0,
                                const float* __restrict__ gsum2,
                                const float* __restrict__ gsum3,
                                float* __restrict__ elig_out) {
  using namespace snn;
  int i = threadIdx.x;            // 256 threads: i = a*64 + h
  int a = i >> 6, h = i & 63;
  const float invB = 1.f / (float)Bsz;
  float e = elig0[i];
  #pragma unroll
  for (int t = 0; t < T; ++t) {
    float post = gsum3[t * A + a] * invB;
    float pre  = gsum2[t * H + h] * invB;
    e = ELIG_DECAY * e + C_STDP * post * pre;
  }
  elig_out[i] = e;
}

extern "C" void kernel_launch(void* const* d_in, const int* in_sizes, int n_in,
                              void* d_out, int out_size, void* d_ws, size_t ws_size,
                              hipStream_t stream) {
  using namespace snn;
  const float* x     = (const float*)d_in[0];
  const float* W1    = (const float*)d_in[1];
  const float* b1    = (const float*)d_in[2];
  const float* W2    = (const float*)d_in[3];
  const float* b2    = (const float*)d_in[4];
  const float* W3    = (const float*)d_in[5];
  const float* b3    = (const float*)d_in[6];
  const float* elig0 = (const float*)d_in[7];

  float* probs    = (float*)d_out;            // [B, A]
  float* elig_out = probs + (size_t)Bsz * A;  // [A, H]
  float* gsum2 = (float*)d_ws;                // [T, H] batch spike sums (s2)
  float* gsum3 = gsum2 + T * H;               // [T, A] batch spike sums (s3)

  snn_zero_sums_kernel<<<1, 1024, 0, stream>>>(gsum2, gsum3);
  snn_forward_kernel<<<Bsz / ROWS_PER_BLOCK, BLOCK, 0, stream>>>(
      x, W1, b1, W2, b2, W3, b3, probs, gsum2, gsum3);
  snn_elig_kernel<<<1, 256, 0, stream>>>(elig0, gsum2, gsum3, elig_out);
}